// MeanStdMemory_26800595927115
// MI455X (gfx1250) — compile-verified
//
#include <hip/hip_runtime.h>
#include <hip/hip_bf16.h>
#include <math.h>

// Problem constants (from reference setup_inputs)
#define NB   16      // batch
#define NN   2048    // n_nodes
#define DIM  256     // feature dim
#define SZ   16384   // codebook size
#define TOPK 50

typedef float v2f __attribute__((ext_vector_type(2)));
typedef float v8f __attribute__((ext_vector_type(8)));

// ---------------------------------------------------------------------------
// Kernel 1: per-batch mean / std over nodes + squared norms of those vectors.
// grid = NB blocks, 256 threads (thread d handles feature dim d).
// ---------------------------------------------------------------------------
__global__ void batch_stats_kernel(const float* __restrict__ x,
                                   float* __restrict__ w_mean,
                                   float* __restrict__ w_std,
                                   float* __restrict__ w_nsM,
                                   float* __restrict__ w_nsS) {
    const int b = blockIdx.x;
    const int d = threadIdx.x;
    const float* p = x + (size_t)b * NN * DIM + d;
    float s = 0.f, s2 = 0.f;
    for (int n = 0; n < NN; ++n) {
        float v = p[(size_t)n * DIM];
        s  += v;
        s2 += v * v;
    }
    const float inv = 1.0f / (float)NN;
    float m  = s * inv;
    float var = fmaxf(s2 * inv - m * m, 0.0f);
    float sd = sqrtf(var);
    w_mean[b * DIM + d] = m;
    w_std [b * DIM + d] = sd;

    __shared__ float red[256];
    // reduce mean^2
    red[d] = m * m;
    __syncthreads();
    for (int off = 128; off > 0; off >>= 1) {
        if (d < off) red[d] += red[d + off];
        __syncthreads();
    }
    if (d == 0) w_nsM[b] = red[0];
    __syncthreads();
    // reduce std^2
    red[d] = sd * sd;
    __syncthreads();
    for (int off = 128; off > 0; off >>= 1) {
        if (d < off) red[d] += red[d + off];
        __syncthreads();
    }
    if (d == 0) w_nsS[b] = red[0];
}

// ---------------------------------------------------------------------------
// Kernel 2: squared row norms of means / stds.  One wave32 per row.
// grid = SZ/8 blocks, 256 threads (8 waves).
// ---------------------------------------------------------------------------
__global__ void row_norms_kernel(const float* __restrict__ means,
                                 const float* __restrict__ stds,
                                 float* __restrict__ rnM,
                                 float* __restrict__ rnS) {
    const int wave = threadIdx.x >> 5;
    const int lane = threadIdx.x & 31;
    const int row  = blockIdx.x * 8 + wave;
    const float* pm = means + (size_t)row * DIM;
    const float* ps = stds  + (size_t)row * DIM;
    float am = 0.f, as = 0.f;
    for (int i = 0; i < DIM / 32; ++i) {
        float vm = pm[lane + i * 32];
        float vs = ps[lane + i * 32];
        am += vm * vm;
        as += vs * vs;
    }
    for (int o = 16; o > 0; o >>= 1) {
        am += __shfl_xor(am, o, 32);
        as += __shfl_xor(as, o, 32);
    }
    if (lane == 0) { rnM[row] = am; rnS[row] = as; }
}

// ---------------------------------------------------------------------------
// Kernel 3: distances via fp32 WMMA.
//   dot[m,n] = means[m,:] . batch_mean[n,:]  (and same for stds)
//   ds[m*16+n] = sqrt(rnM[m] - 2*dotM + nsM[n]) + sqrt(rnS[m] - 2*dotS + nsS[n])
// One wave handles a 16-row tile of m, full N=16 batches.
// grid = SZ/16/8 = 128 blocks, 256 threads (8 waves).
//
// V_WMMA_F32_16X16X4_F32 operand layout (ISA 7.12.2):
//   A (16x4, MxK):  lane = M + 16*h  holds K = kb+2h (VGPR0), kb+2h+1 (VGPR1)
//   B (4x16, KxN):  lane = N + 16*h  holds K rows kb+2h, kb+2h+1
//   C/D (16x16):    VGPR r -> M = r + 8*h, N = lane&15
// ---------------------------------------------------------------------------
__global__ void wmma_dist_kernel(const float* __restrict__ means,
                                 const float* __restrict__ stds,
                                 const float* __restrict__ bmean,  // [NB][DIM]
                                 const float* __restrict__ bstd,   // [NB][DIM]
                                 const float* __restrict__ rnM,
                                 const float* __restrict__ rnS,
                                 const float* __restrict__ nsM,
                                 const float* __restrict__ nsS,
                                 float* __restrict__ ds) {          // [SZ][NB]
    const int waveG  = blockIdx.x * 8 + (threadIdx.x >> 5);
    const int s_base = waveG * 16;
    const int lane   = threadIdx.x & 31;
    const int half   = lane >> 4;       // 0 or 1
    const int mrow   = lane & 15;       // M index for A, N index for B/C/D
    const int koff   = half * 2;

    const float* aM_row = means + (size_t)(s_base + mrow) * DIM;
    const float* aS_row = stds  + (size_t)(s_base + mrow) * DIM;
    const float* bM_row = bmean + (size_t)mrow * DIM;   // batch n = mrow
    const float* bS_row = bstd  + (size_t)mrow * DIM;

    v8f accM = {};
    v8f accS = {};
    for (int kb = 0; kb < DIM; kb += 4) {
        v2f aM = *(const v2f*)(aM_row + kb + koff);
        v2f bM = *(const v2f*)(bM_row + kb + koff);
        v2f aS = *(const v2f*)(aS_row + kb + koff);
        v2f bS = *(const v2f*)(bS_row + kb + koff);
        accM = __builtin_amdgcn_wmma_f32_16x16x4_f32(
            false, aM, false, bM, (short)0, accM, false, false);
        accS = __builtin_amdgcn_wmma_f32_16x16x4_f32(
            false, aS, false, bS, (short)0, accS, false, false);
    }

    const int n = mrow;                 // batch index for this lane's C column
    const float nm = nsM[n];
    const float ns = nsS[n];
    for (int r = 0; r < 8; ++r) {
        int m_g = s_base + r + 8 * half;
        float dm = rnM[m_g] - 2.0f * accM[r] + nm;
        float dsv = rnS[m_g] - 2.0f * accS[r] + ns;
        float dist = sqrtf(fmaxf(dm, 0.0f)) + sqrtf(fmaxf(dsv, 0.0f));
        ds[(size_t)m_g * NB + n] = dist;
    }
}

// ---------------------------------------------------------------------------
// Kernel 4: top-50 (smallest, lowest-index tie-break), softmax-of-exp weights,
// goal vectors, then collapse to per-(b,d) scale/offset.
// grid = NB blocks, 256 threads.  Destroys ds (rewritten next call).
// ---------------------------------------------------------------------------
__global__ void topk_combine_kernel(const float* __restrict__ means,
                                    const float* __restrict__ stds,
                                    const float* __restrict__ bmean,
                                    const float* __restrict__ bstd,
                                    float* __restrict__ ds,        // [SZ][NB]
                                    const float* __restrict__ t1p,
                                    const float* __restrict__ t2p,
                                    float* __restrict__ w_scale,
                                    float* __restrict__ w_offset) {
    const int b   = blockIdx.x;
    const int tid = threadIdx.x;

    __shared__ float sval[256];
    __shared__ int   sidx[256];
    __shared__ int   selIdx[TOPK];
    __shared__ float selD[TOPK];
    __shared__ float wts[TOPK];

    // --- iterative argmin x50 ---
    for (int it = 0; it < TOPK; ++it) {
        float best = INFINITY;
        int   bidx = 0x7fffffff;
        for (int j = tid; j < SZ; j += 256) {
            float v = ds[(size_t)j * NB + b];
            if (v < best || (v == best && j < bidx)) { best = v; bidx = j; }
        }
        sval[tid] = best; sidx[tid] = bidx;
        __syncthreads();
        for (int off = 128; off > 0; off >>= 1) {
            if (tid < off) {
                float v = sval[tid + off]; int ix = sidx[tid + off];
                if (v < sval[tid] || (v == sval[tid] && ix < sidx[tid])) {
                    sval[tid] = v; sidx[tid] = ix;
                }
            }
            __syncthreads();
        }
        if (tid == 0) {
            selIdx[it] = sidx[0];
            selD[it]   = sval[0];
            ds[(size_t)sidx[0] * NB + b] = INFINITY;  // exclude from next pass
        }
        __syncthreads();
    }

    // --- weights: s_i = exp(-t1*d_i); w = softmax(s) ---
    if (tid == 0) {
        float t1 = t1p[0];
        float mx = -INFINITY;
        for (int i = 0; i < TOPK; ++i) {
            float si = expf(-t1 * selD[i]);
            wts[i] = si;
            mx = fmaxf(mx, si);
        }
        float sum = 0.f;
        for (int i = 0; i < TOPK; ++i) {
            float e = expf(wts[i] - mx);
            wts[i] = e;
            sum += e;
        }
        float isum = 1.0f / sum;
        for (int i = 0; i < TOPK; ++i) wts[i] *= isum;
    }
    __syncthreads();

    // --- goal vectors and per-dim affine collapse ---
    const int d = tid;
    float mg = 0.f, sg = 0.f;
    for (int i = 0; i < TOPK; ++i) {
        int   idx = selIdx[i];
        float wv  = wts[i];
        mg = fmaf(wv, means[(size_t)idx * DIM + d], mg);
        sg = fmaf(wv, stds [(size_t)idx * DIM + d], sg);
    }
    float t2   = t2p[0];
    float lerp = 1.0f / (1.0f + expf(-t2));
    float bm = bmean[b * DIM + d];
    float bs = bstd [b * DIM + d];
    float mean_final = lerp * mg + (1.0f - lerp) * bm;
    float std_final  = lerp * sg + (1.0f - lerp) * bs;
    float sc  = std_final / bs;
    float off = mean_final - sc * bm;
    w_scale [b * DIM + d] = sc;
    w_offset[b * DIM + d] = off;
}

// ---------------------------------------------------------------------------
// Kernel 5: out = scale*x + offset, float4 vectorized.
// grid = NB*NN*DIM/4/256 = 8192 blocks.
// ---------------------------------------------------------------------------
__global__ void transform_kernel(const float* __restrict__ x,
                                 const float* __restrict__ w_scale,
                                 const float* __restrict__ w_offset,
                                 float* __restrict__ out) {
    const size_t i    = (size_t)blockIdx.x * blockDim.x + threadIdx.x;
    const size_t base = i * 4;                 // element index
    const int b  = (int)(base >> 19);          // NN*DIM = 2^19 per batch
    const int d0 = (int)(base & (DIM - 1));    // dim of first of the 4
    float4 xv  = *(const float4*)(x + base);
    float4 scv = *(const float4*)(w_scale  + b * DIM + d0);
    float4 ofv = *(const float4*)(w_offset + b * DIM + d0);
    float4 o;
    o.x = fmaf(scv.x, xv.x, ofv.x);
    o.y = fmaf(scv.y, xv.y, ofv.y);
    o.z = fmaf(scv.z, xv.z, ofv.z);
    o.w = fmaf(scv.w, xv.w, ofv.w);
    *(float4*)(out + base) = o;
}

// ---------------------------------------------------------------------------
extern "C" void kernel_launch(void* const* d_in, const int* in_sizes, int n_in,
                              void* d_out, int out_size, void* d_ws, size_t ws_size,
                              hipStream_t stream) {
    const float* node_fts = (const float*)d_in[0];  // [16,2048,256]
    const float* means    = (const float*)d_in[1];  // [16384,256]
    const float* stds     = (const float*)d_in[2];  // [16384,256]
    const float* temp1    = (const float*)d_in[3];  // scalar
    const float* temp2    = (const float*)d_in[4];  // scalar
    float* out = (float*)d_out;

    // workspace layout (float offsets)
    float* ws       = (float*)d_ws;
    float* w_mean   = ws;                         // NB*DIM       = 4096
    float* w_std    = ws + 4096;                  // NB*DIM       = 4096
    float* w_nsM    = ws + 8192;                  // NB           = 16
    float* w_nsS    = ws + 8208;                  // NB           = 16
    float* w_rnM    = ws + 8224;                  // SZ           = 16384
    float* w_rnS    = ws + 24608;                 // SZ           = 16384
    float* w_ds     = ws + 40992;                 // SZ*NB        = 262144
    float* w_scale  = ws + 303136;                // NB*DIM       = 4096
    float* w_offset = ws + 307232;                // NB*DIM       = 4096
    (void)in_sizes; (void)n_in; (void)out_size; (void)ws_size;

    batch_stats_kernel<<<NB, 256, 0, stream>>>(node_fts, w_mean, w_std, w_nsM, w_nsS);
    row_norms_kernel<<<SZ / 8, 256, 0, stream>>>(means, stds, w_rnM, w_rnS);
    wmma_dist_kernel<<<SZ / 16 / 8, 256, 0, stream>>>(means, stds, w_mean, w_std,
                                                      w_rnM, w_rnS, w_nsM, w_nsS, w_ds);
    topk_combine_kernel<<<NB, 256, 0, stream>>>(means, stds, w_mean, w_std, w_ds,
                                                temp1, temp2, w_scale, w_offset);
    transform_kernel<<<(NB * NN * DIM) / 4 / 256, 256, 0, stream>>>(node_fts, w_scale,
                                                                    w_offset, out);
}